// MultiHeadAttentionLayer_4080218931583
// MI455X (gfx1250) — compile-verified
//
#include <hip/hip_runtime.h>
#include <hip/hip_bf16.h>
#include <math.h>

// ---------------------------------------------------------------------------
// Graph multi-head attention (DGL-style GraphConv Q/K/V + edge softmax-ish)
//   N = n_nodes, F = 128 feats, H = 8 heads, Dh = 16, E = n_edges
// ---------------------------------------------------------------------------

typedef float v2f __attribute__((ext_vector_type(2)));
typedef float v8f __attribute__((ext_vector_type(8)));

#define FEAT 128
#define HEADS 8
#define DHEAD 16

// -------------------- degree accumulation --------------------
__global__ void deg_kernel(const int* __restrict__ src, const int* __restrict__ dst,
                           float* __restrict__ deg_out, float* __restrict__ deg_in, int E) {
    int e = blockIdx.x * blockDim.x + threadIdx.x;
    if (e < E) {
        atomicAdd(&deg_out[src[e]], 1.0f);
        atomicAdd(&deg_in[dst[e]], 1.0f);
    }
}

// deg -> rsqrt(max(deg,1)) in place, for both arrays
__global__ void norm_kernel(float* __restrict__ a, float* __restrict__ b, int N) {
    int i = blockIdx.x * blockDim.x + threadIdx.x;
    if (i < N) {
        a[i] = rsqrtf(fmaxf(a[i], 1.0f));
        b[i] = rsqrtf(fmaxf(b[i], 1.0f));
    }
}

// -------------------- WMMA GEMM: X = (h * norm_src[:,None]) @ W --------------------
// block: 256 threads (8 waves). Block computes rows [rowBase, rowBase+16) x all 128 cols.
// wave w computes 16x16 tile at colBase = w*16 using V_WMMA_F32_16X16X4_F32, K-loop of 32.
__global__ void gemm_wmma_kernel(const float* __restrict__ h,
                                 const float* __restrict__ norm_src,
                                 const float* __restrict__ W,
                                 float* __restrict__ X, int N) {
    __shared__ float As[16][FEAT + 4];   // +4 pad -> conflict-free fragment reads

    const int tid     = threadIdx.x;
    const int rowBase = blockIdx.x * 16;

    // cooperative load of the 16x128 A tile, scaled by norm_src
    #pragma unroll
    for (int i = 0; i < 8; ++i) {
        int idx = tid + i * 256;          // 0..2047
        int r = idx >> 7;                 // 0..15
        int c = idx & (FEAT - 1);
        int gr = rowBase + r;
        if (gr >= N) gr = N - 1;          // clamp (stores are guarded)
        As[r][c] = h[gr * FEAT + c] * norm_src[gr];
    }
    __syncthreads();

    const int lane    = tid & 31;
    const int wave    = tid >> 5;        // 0..7
    const int colBase = wave * 16;
    const int mrow    = lane & 15;       // row (A) / col (B) within tile
    const int khalf   = (lane >> 4) * 2; // lanes 0-15 -> K0/K1, lanes 16-31 -> K2/K3

    v8f acc = {};
    #pragma unroll 4
    for (int k = 0; k < FEAT; k += 4) {
        v2f a, b;
        a.x = As[mrow][k + khalf];
        a.y = As[mrow][k + khalf + 1];
        b.x = W[(k + khalf)     * FEAT + colBase + mrow];
        b.y = W[(k + khalf + 1) * FEAT + colBase + mrow];
        acc = __builtin_amdgcn_wmma_f32_16x16x4_f32(
            /*neg_a=*/false, a, /*neg_b=*/false, b,
            /*c_mod=*/(short)0, acc, /*reuse_a=*/false, /*reuse_b=*/false);
    }

    // C/D layout: VGPR r -> M = r (lanes 0-15) or r+8 (lanes 16-31); N = lane&15
    const int rOff = (lane >> 4) << 3;
    if (rowBase + 16 <= N) {
        // full tile (common case): single scalar branch, 8 straight stores
        float* Xb = X + (rowBase + rOff) * FEAT + colBase + (lane & 15);
        #pragma unroll
        for (int r = 0; r < 8; ++r)
            Xb[r * FEAT] = acc[r];
    } else {
        #pragma unroll
        for (int r = 0; r < 8; ++r) {
            int outRow = rowBase + r + rOff;
            if (outRow < N) X[outRow * FEAT + colBase + (lane & 15)] = acc[r];
        }
    }
}

// -------------------- edge scatter: agg[dst] += X[src] --------------------
// one thread handles 4 consecutive floats of one edge (32 threads / edge)
__global__ void scatter_kernel(const float* __restrict__ X,
                               const int* __restrict__ src, const int* __restrict__ dst,
                               float* __restrict__ agg, int E) {
    int tid = blockIdx.x * blockDim.x + threadIdx.x;
    int e = tid >> 5;
    if (e >= E) return;
    int lane = tid & 31;
    int s = src[e], d = dst[e];
    const float4 v = *reinterpret_cast<const float4*>(X + s * FEAT + lane * 4);
    float* out = agg + d * FEAT + lane * 4;
    atomicAdd(out + 0, v.x);
    atomicAdd(out + 1, v.y);
    atomicAdd(out + 2, v.z);
    atomicAdd(out + 3, v.w);
}

// -------------------- bias + relu (in place on agg buffers) --------------------
__global__ void bias_relu_kernel(float* __restrict__ Q, float* __restrict__ K, float* __restrict__ V,
                                 const float* __restrict__ bQ, const float* __restrict__ bK,
                                 const float* __restrict__ bV,
                                 const float* __restrict__ norm_dst, int N) {
    int i = blockIdx.x * blockDim.x + threadIdx.x;
    if (i >= N * FEAT) return;
    int node = i >> 7;
    int c = i & (FEAT - 1);
    float nd = norm_dst[node];
    Q[i] = fmaxf(Q[i] * nd + bQ[c], 0.0f);
    K[i] = fmaxf(K[i] * nd + bK[c], 0.0f);
    V[i] = fmaxf(V[i] * nd + bV[c], 0.0f);
}

// -------------------- wave-per-edge attention --------------------
// lane l handles floats [l*4, l*4+4) of the 128-float row; head = l/4.
// score_h = exp(clip(dot(K[src],Q[dst])_h / 4, -10, 10))
__global__ void attn_kernel(const float* __restrict__ Q, const float* __restrict__ K,
                            const float* __restrict__ V,
                            const int* __restrict__ src, const int* __restrict__ dst,
                            float* __restrict__ out, float* __restrict__ z, int E) {
    int tid = blockIdx.x * blockDim.x + threadIdx.x;
    int e = tid >> 5;
    if (e >= E) return;
    int lane = tid & 31;
    int s = src[e], d = dst[e];

    const float4 kv = *reinterpret_cast<const float4*>(K + s * FEAT + lane * 4);
    const float4 qv = *reinterpret_cast<const float4*>(Q + d * FEAT + lane * 4);
    float p = kv.x * qv.x + kv.y * qv.y + kv.z * qv.z + kv.w * qv.w;
    // reduce within groups of 4 lanes (one head each)
    p += __shfl_xor(p, 1);
    p += __shfl_xor(p, 2);
    float sc = __expf(fminf(fmaxf(p * 0.25f, -10.0f), 10.0f));

    const float4 vv = *reinterpret_cast<const float4*>(V + s * FEAT + lane * 4);
    float* o = out + d * FEAT + lane * 4;
    atomicAdd(o + 0, vv.x * sc);
    atomicAdd(o + 1, vv.y * sc);
    atomicAdd(o + 2, vv.z * sc);
    atomicAdd(o + 3, vv.w * sc);
    if ((lane & 3) == 0) atomicAdd(&z[d * HEADS + (lane >> 2)], sc);
}

// -------------------- final divide --------------------
__global__ void finalize_kernel(float* __restrict__ out, const float* __restrict__ z, int N) {
    int i = blockIdx.x * blockDim.x + threadIdx.x;
    if (i >= N * FEAT) return;
    int node = i >> 7;
    int head = (i >> 4) & (HEADS - 1);
    out[i] = out[i] / (z[node * HEADS + head] + 1e-6f);
}

// ---------------------------------------------------------------------------
extern "C" void kernel_launch(void* const* d_in, const int* in_sizes, int n_in,
                              void* d_out, int out_size, void* d_ws, size_t ws_size,
                              hipStream_t stream) {
    const float* h   = (const float*)d_in[0];
    const float* W_Q = (const float*)d_in[1];
    const float* b_Q = (const float*)d_in[2];
    const float* W_K = (const float*)d_in[3];
    const float* b_K = (const float*)d_in[4];
    const float* W_V = (const float*)d_in[5];
    const float* b_V = (const float*)d_in[6];
    const int*   src = (const int*)d_in[7];
    const int*   dst = (const int*)d_in[8];

    const int N = in_sizes[0] / FEAT;
    const int E = in_sizes[7];

    // workspace layout (floats):
    //   norm_src[N] | norm_dst[N] | z[8N] | Q[128N] | K[128N] | V[128N] | Xtmp[128N]
    float* ws       = (float*)d_ws;
    float* norm_src = ws;
    float* norm_dst = ws + (size_t)N;
    float* zbuf     = ws + (size_t)2 * N;
    float* Qb       = ws + (size_t)10 * N;
    float* Kb       = Qb + (size_t)FEAT * N;
    float* Vb       = Kb + (size_t)FEAT * N;
    float* Xtmp     = Vb + (size_t)FEAT * N;

    float* out = (float*)d_out;

    // zero: norms + z + Q/K/V agg buffers (contiguous) and d_out
    hipMemsetAsync(d_ws, 0, (size_t)(10 + 3 * FEAT) * N * sizeof(float), stream);
    hipMemsetAsync(d_out, 0, (size_t)N * FEAT * sizeof(float), stream);

    const int T = 256;

    // 1. degrees -> norms
    deg_kernel<<<(E + T - 1) / T, T, 0, stream>>>(src, dst, norm_src, norm_dst, E);
    norm_kernel<<<(N + T - 1) / T, T, 0, stream>>>(norm_src, norm_dst, N);

    // 2. three graph convs: WMMA GEMM then edge scatter
    const int gemmBlocks    = (N + 15) / 16;
    const int scatterBlocks = ((E * 32) + T - 1) / T;

    gemm_wmma_kernel<<<gemmBlocks, T, 0, stream>>>(h, norm_src, W_Q, Xtmp, N);
    scatter_kernel<<<scatterBlocks, T, 0, stream>>>(Xtmp, src, dst, Qb, E);

    gemm_wmma_kernel<<<gemmBlocks, T, 0, stream>>>(h, norm_src, W_K, Xtmp, N);
    scatter_kernel<<<scatterBlocks, T, 0, stream>>>(Xtmp, src, dst, Kb, E);

    gemm_wmma_kernel<<<gemmBlocks, T, 0, stream>>>(h, norm_src, W_V, Xtmp, N);
    scatter_kernel<<<scatterBlocks, T, 0, stream>>>(Xtmp, src, dst, Vb, E);

    // 3. bias + relu with dst-side norm
    bias_relu_kernel<<<((N * FEAT) + T - 1) / T, T, 0, stream>>>(
        Qb, Kb, Vb, b_Q, b_K, b_V, norm_dst, N);

    // 4. edge attention (wave per edge) accumulating into out / z
    attn_kernel<<<scatterBlocks, T, 0, stream>>>(Qb, Kb, Vb, src, dst, out, zbuf, E);

    // 5. normalize
    finalize_kernel<<<((N * FEAT) + T - 1) / T, T, 0, stream>>>(out, zbuf, N);
}